// RepulsionAttentionModel_50122268344505
// MI455X (gfx1250) — compile-verified
//
#include <hip/hip_runtime.h>
#include <math.h>

#define TS    1024      // sequence length S
#define BLOCK 256       // 8 waves (wave32)
#define QPB   128       // queries per block: 8 waves * 16 rows

typedef __attribute__((ext_vector_type(2))) float v2f;
typedef __attribute__((ext_vector_type(8))) float v8f;

#define LOG2E      1.4426950408889634f
#define LN2        0.6931471805599453f
#define EPSF       1e-8f
#define PHI_HALF   0.8090169943749475f     // PHI/2
#define LN_R_MIN_C -0.9624236501192069f    // ln(1 - 1/phi)
#define ZMAX_C     0.6180339887498949f     // 1 - GAP = 1/phi
#define TWO_PI_C   6.2831853071795865f
#define INV_2PI    0.15915494309189535f
#define DT_C       0.01f

__device__ __forceinline__ float fexp2(float x){ return __builtin_amdgcn_exp2f(x); }
__device__ __forceinline__ float flog2(float x){ return __builtin_amdgcn_logf(x); }
__device__ __forceinline__ float frcp (float x){ return __builtin_amdgcn_rcpf(x); }
__device__ __forceinline__ float frsq (float x){ return __builtin_amdgcn_rsqf(x); }
__device__ __forceinline__ float fsqrtx(float x){ return __builtin_amdgcn_sqrtf(x); }

// ---------------------------------------------------------------------------
// Prep: gather token states, derive (x,y,z,n) for the fixed initial set into
// a per-batch packed layout [b][ X(1024) | Y(1024) | Z(1024) | N(1024) ] so a
// step block can DMA its whole 16KB tile to LDS with async loads, and seed the
// raw (ln_r, theta, z) ping buffer with the initial states.
// ---------------------------------------------------------------------------
__global__ void prep_kernel(const int* __restrict__ ids,
                            const float* __restrict__ tstates,
                            float* __restrict__ gInit,
                            float* __restrict__ gLr, float* __restrict__ gTh,
                            float* __restrict__ gZz, int n)
{
    int p = blockIdx.x * blockDim.x + threadIdx.x;
    if (p >= n) return;
    int id = ids[p];
    float lr = tstates[id*3+0];
    float th = tstates[id*3+1];
    float zz = tstates[id*3+2];
    float r  = fexp2(lr * LOG2E);
    float x  = r * __cosf(th);
    float y  = r * __sinf(th);
    int   b  = p >> 10;
    int   s  = p & (TS - 1);
    float* base = gInit + (b << 12);       // 4096 floats per batch
    base[0*TS + s] = x;
    base[1*TS + s] = y;
    base[2*TS + s] = zz;
    base[3*TS + s] = x*x + y*y + zz*zz;
    gLr[p] = lr; gTh[p] = th; gZz[p] = zz;
}

// ---------------------------------------------------------------------------
// One integration step. Pairwise q = n_i + n_j - 2<s_i,s_j> via WMMA 16x16x4:
//   A_i = (-2x,-2y,-2z,1)   B_j = (x,y,z,n_j)   =>  C = -2<s_i,s_j> + n_j
// Forces factorized: F_i = (Sum g)*s_i - Sum g*s_j.
// Fixed-set tile staged via GLOBAL_LOAD_ASYNC_TO_LDS_B128 (ASYNCcnt),
// overlapped with the VALU derivation of the current set.
// ---------------------------------------------------------------------------
__global__ void __launch_bounds__(BLOCK)
step_kernel(const float* __restrict__ gInit,
            const float* __restrict__ inLr, const float* __restrict__ inTh,
            const float* __restrict__ inZz,
            float* __restrict__ outLr, float* __restrict__ outTh,
            float* __restrict__ outZz,
            const float* __restrict__ temp_p, const float* __restrict__ pw_p,
            const float* __restrict__ cw_p,   const float* __restrict__ fw_p,
            float sinPast, float sinPres, float sinFut,
            float* __restrict__ finalOut, int isFinal)
{
    __shared__ __align__(16) float sInit[4*TS];   // fixed set: X|Y|Z|N
    __shared__ __align__(16) float sCur [4*TS];   // current set: X|Y|Z|N

    float* sX0 = sInit;            float* sY0 = sInit + TS;
    float* sZ0 = sInit + 2*TS;     float* sN0 = sInit + 3*TS;
    float* sXc = sCur;             float* sYc = sCur + TS;
    float* sZc = sCur + 2*TS;      float* sNc = sCur + 3*TS;

    const int bpb   = TS / QPB;                 // blocks per batch (8)
    const int b     = blockIdx.x / bpb;
    const int qb    = (blockIdx.x % bpb) * QPB; // query tile base in batch
    const int pbase = b * TS;
    const int lane  = (int)(threadIdx.x & 31u);
    const int wave  = (int)(threadIdx.x >> 5);
    const int i0    = qb + wave * 16;           // this wave's 16-row i-tile
    const int m     = lane & 15;
    const int half  = lane >> 4;
    const int row8  = half << 3;

    // --- Kick off async DMA of the 16KB fixed-set tile: global -> LDS.
    {
        unsigned ldsbase = (unsigned)(size_t)(void*)sInit;   // LDS aperture: low 32 bits
        unsigned gbatch  = (unsigned)(b << 14);              // 16384 bytes per batch
        #pragma unroll
        for (int it = 0; it < (4*TS*4) / (BLOCK*16); ++it) { // 4 sweeps of 4KB
            unsigned off  = (unsigned)((it*BLOCK + (int)threadIdx.x) * 16);
            unsigned ldsa = ldsbase + off;
            unsigned voff = gbatch + off;
            asm volatile("global_load_async_to_lds_b128 %0, %1, %2"
                         :: "v"(ldsa), "v"(voff), "s"(gInit)
                         : "memory");
        }
    }

    // --- Meanwhile derive the current set (exp/cos/sin per point) into LDS.
    for (int t = (int)threadIdx.x; t < TS; t += BLOCK) {
        float lr = inLr[pbase + t], th = inTh[pbase + t], zz = inZz[pbase + t];
        float r = fexp2(lr * LOG2E);
        float x = r * __cosf(th);
        float y = r * __sinf(th);
        sXc[t] = x; sYc[t] = y; sZc[t] = zz;
        sNc[t] = x*x + y*y + zz*zz;
    }

    asm volatile("s_wait_asynccnt 0x0" ::: "memory");
    __syncthreads();

    const float temp = temp_p[0];
    const float KEXP = -0.5f * LOG2E * frcp(temp);           // exp(-u^2/2T) = exp2(u^2*KEXP)
    const float wa = (pw_p[0]*sinPast + fw_p[0]*sinFut) * (1.0f/(float)TS);
    const float wc = (cw_p[0]*sinPres) * (1.0f/(float)TS);

    // A-matrix (16x4 f32): lanes 0-15 hold K=0,1; lanes 16-31 hold K=2,3.
    v2f Av;
    Av.x = half ? (-2.0f * sZc[i0 + m]) : (-2.0f * sXc[i0 + m]);
    Av.y = half ? 1.0f                  : (-2.0f * sYc[i0 + m]);

    // Row constants per accumulator VGPR k: row M = k + 8*half (C/D layout).
    float rowNE[8], rowXC[8];
    #pragma unroll
    for (int k = 0; k < 8; ++k) {
        int rr = i0 + k + row8;
        rowNE[k] = sNc[rr] + EPSF;          // n_i + EPS
        rowXC[k] = sXc[rr] + PHI_HALF;      // x_i + phi/2
    }

    float accS[8], accX[8], accY[8], accZ[8];
    #pragma unroll
    for (int k = 0; k < 8; ++k) { accS[k]=0.f; accX[k]=0.f; accY[k]=0.f; accZ[k]=0.f; }

    const v8f Cz = {};

    for (int j0 = 0; j0 < TS; j0 += 16) {
        const int jn = j0 + m;              // column N = lane&15 (both halves)

        // ---- term vs fixed initial states (weight wa = w_past + w_future)
        {
            float xc = sX0[jn], yc = sY0[jn], zc = sZ0[jn], nc = sN0[jn];
            v2f Bv; Bv.x = half ? zc : xc; Bv.y = half ? nc : yc;
            v8f C = __builtin_amdgcn_wmma_f32_16x16x4_f32(
                        false, Av, false, Bv, (short)0, Cz, false, false);
            #pragma unroll
            for (int k = 0; k < 8; ++k) {
                float q  = fmaxf(C[k] + rowNE[k], EPSF);
                float d  = fsqrtx(q);
                float den = (d*q + EPSF) * d;           // (d^3 + EPS) * d
                float u  = rowXC[k] - yc;
                float wg = wa * fexp2(u*u*KEXP) * frcp(den);
                accS[k] += wg;
                accX[k] += wg * xc;
                accY[k] += wg * yc;
                accZ[k] += wg * zc;
            }
        }
        // ---- term vs current states (weight wc = w_present)
        {
            float xc = sXc[jn], yc = sYc[jn], zc = sZc[jn], nc = sNc[jn];
            v2f Bv; Bv.x = half ? zc : xc; Bv.y = half ? nc : yc;
            v8f C = __builtin_amdgcn_wmma_f32_16x16x4_f32(
                        false, Av, false, Bv, (short)0, Cz, false, false);
            #pragma unroll
            for (int k = 0; k < 8; ++k) {
                float q  = fmaxf(C[k] + rowNE[k], EPSF);
                float d  = fsqrtx(q);
                float den = (d*q + EPSF) * d;
                float u  = rowXC[k] - yc;
                float wg = wc * fexp2(u*u*KEXP) * frcp(den);
                accS[k] += wg;
                accX[k] += wg * xc;
                accY[k] += wg * yc;
                accZ[k] += wg * zc;
            }
        }
    }

    // Reduce over the 16 columns held in each half-wave.
    #pragma unroll
    for (int k = 0; k < 8; ++k) {
        #pragma unroll
        for (int msk = 1; msk < 16; msk <<= 1) {
            accS[k] += __shfl_xor(accS[k], msk, 32);
            accX[k] += __shfl_xor(accX[k], msk, 32);
            accY[k] += __shfl_xor(accY[k], msk, 32);
            accZ[k] += __shfl_xor(accZ[k], msk, 32);
        }
    }

    // Lane (m==k) of each half owns row k + 8*half: integrate + bounds + born.
    #pragma unroll
    for (int k = 0; k < 8; ++k) {
        if (m == k) {
            int row = i0 + k + row8;
            int p   = pbase + row;
            float Fx = accS[k]*sXc[row] - accX[k];
            float Fy = accS[k]*sYc[row] - accY[k];
            float Fz = accS[k]*sZc[row] - accZ[k];

            float lr = inLr[p] + DT_C * Fx;
            lr = fminf(fmaxf(lr, LN_R_MIN_C), 0.0f);
            float th = inTh[p] + DT_C * Fy;
            th = th - floorf(th * INV_2PI) * TWO_PI_C;      // mod 2pi -> [0,2pi)
            float zz = inZz[p] + DT_C * Fz;
            zz = fminf(fmaxf(zz, 0.0f), ZMAX_C);

            float r  = fexp2(lr * LOG2E);
            float rs = frsq(r*r + zz*zz + EPSF);            // 1/norm
            float ol = flog2(r*rs + EPSF) * LN2;
            float oz = zz * rs;

            outLr[p] = ol; outTh[p] = th; outZz[p] = oz;
            if (isFinal) {
                finalOut[p*3+0] = ol;
                finalOut[p*3+1] = th;
                finalOut[p*3+2] = oz;
            }
        }
    }
}

// ---------------------------------------------------------------------------
extern "C" void kernel_launch(void* const* d_in, const int* in_sizes, int n_in,
                              void* d_out, int out_size, void* d_ws, size_t ws_size,
                              hipStream_t stream)
{
    const int*   ids     = (const int*)  d_in[0];   // (B,S) token ids
    const float* tstates = (const float*)d_in[1];   // (V,3)
    const float* temp_p  = (const float*)d_in[2];
    const float* pw_p    = (const float*)d_in[3];
    const float* cw_p    = (const float*)d_in[4];
    const float* fw_p    = (const float*)d_in[5];

    const int BS = in_sizes[0];                     // B*S = 4096

    // Workspace layout: packed init (4*BS) | cur ping (3*BS) | cur pong (3*BS)
    float* w = (float*)d_ws;
    float* gInit = w;
    float* LrA = w + 4*BS;  float* ThA = w + 5*BS;  float* ZzA = w + 6*BS;
    float* LrB = w + 7*BS;  float* ThB = w + 8*BS;  float* ZzB = w + 9*BS;

    prep_kernel<<<(BS + BLOCK - 1)/BLOCK, BLOCK, 0, stream>>>(
        ids, tstates, gInit, LrA, ThA, ZzA, BS);

    const double TWOPI = 6.283185307179586476925286766559;
    float* out = (float*)d_out;

    for (int s = 0; s < 3; ++s) {
        double phase = (double)s * TWOPI / 3.0;
        float sp = (float)sin(phase + TWOPI/3.0);        // w_past factor
        float sc = (float)sin(phase + 2.0*TWOPI/3.0);    // w_present factor
        float sf = (float)sin(phase);                    // w_future factor

        const float* iLr = (s & 1) ? LrB : LrA;
        const float* iTh = (s & 1) ? ThB : ThA;
        const float* iZz = (s & 1) ? ZzB : ZzA;
        float* oLr = (s & 1) ? LrA : LrB;
        float* oTh = (s & 1) ? ThA : ThB;
        float* oZz = (s & 1) ? ZzA : ZzB;

        step_kernel<<<BS / QPB, BLOCK, 0, stream>>>(
            gInit,
            iLr, iTh, iZz, oLr, oTh, oZz,
            temp_p, pw_p, cw_p, fw_p,
            sp, sc, sf,
            out, (s == 2) ? 1 : 0);
    }
}